// traditional_jastrow_9191230013772
// MI455X (gfx1250) — compile-verified
//
#include <hip/hip_runtime.h>

typedef float v2f __attribute__((ext_vector_type(2)));
typedef float v8f __attribute__((ext_vector_type(8)));

#define N_ELEC 32   // N_UP + N_DOWN
#define N_NUC  8
#define N_UP   16

// One wave32 per batch sample. 8 waves (256 threads) per block.
__global__ __launch_bounds__(256)
void jastrow_kernel(const float* __restrict__ elec,     // [B, 32, 32]
                    const float* __restrict__ nuc,      // [B, 32, 8]
                    const float* __restrict__ alpha,    // [8]
                    const float* __restrict__ beta,     // [3, 8]
                    const float* __restrict__ Z,        // [8]
                    const float* __restrict__ L_chi,    // [3]
                    const float* __restrict__ L_u_p,    // [1]
                    const int*   __restrict__ type_idx, // [8]
                    float* __restrict__ out,            // [B]
                    int Bnum)
{
    const int lane = threadIdx.x & 31;
    const int wave = threadIdx.x >> 5;
    const int b    = blockIdx.x * 8 + wave;
    if (b >= Bnum) return;                  // wave-uniform: EXEC stays full

    // ---------- issue all global loads up front (pure b128, coalesced) ----------
    // e-n block: lane = electron index, 8 contiguous floats = 2 x float4
    const float4* n4 = reinterpret_cast<const float4*>(
        nuc + (size_t)b * (N_ELEC * N_NUC) + lane * N_NUC);
    float4 nv0 = n4[0];
    float4 nv1 = n4[1];

    // e-e tile: 8 x float4 per lane covers the whole 32x32 row-major tile
    const float4* e4 = reinterpret_cast<const float4*>(
        elec + (size_t)b * (N_ELEC * N_ELEC));
    float4 ev[8];
    #pragma unroll
    for (int t = 0; t < 8; ++t) ev[t] = e4[t * 32 + lane];

    // ---------- u-term: electron-electron ----------
    const float Lu = L_u_p[0];
    const float a0 = alpha[0];
    float ac[7];                            // alpha[1..7] = coeffs of r^2..r^8
    #pragma unroll
    for (int k = 0; k < 7; ++k) ac[k] = alpha[k + 1];
    const float negInvLu3 = -1.0f / (Lu * Lu * Lu);  // G/(-Lu)^3 = G * negInvLu3
    const float linBase   = 3.0f * a0 / Lu;

    float u_acc = 0.0f;
    const int i_off = lane >> 3;            // row sub-offset for this lane
    const int j0    = (lane & 7) * 4;       // first column for this lane
    #pragma unroll
    for (int t = 0; t < 8; ++t) {
        const int  i    = t * 4 + i_off;
        const bool i_up = (i < N_UP);
        const float rv[4] = {ev[t].x, ev[t].y, ev[t].z, ev[t].w};
        #pragma unroll
        for (int e = 0; e < 4; ++e) {
            const int   j = j0 + e;
            const float r = rv[e];
            const float G = (((j < N_UP)) == i_up) ? 0.25f : 0.5f;
            const float lin = fmaf(G, negInvLu3, linBase);
            float h = ac[6];                // Horner: r^8 down to r^0
            h = fmaf(h, r, ac[5]);
            h = fmaf(h, r, ac[4]);
            h = fmaf(h, r, ac[3]);
            h = fmaf(h, r, ac[2]);
            h = fmaf(h, r, ac[1]);
            h = fmaf(h, r, ac[0]);
            h = fmaf(h, r, lin);
            h = fmaf(h, r, a0);
            const float d   = r - Lu;       // env = (r-Lu)^3 * [r < Lu]
            const float val = d * d * d * h;
            u_acc += ((j > i) && (r < Lu)) ? val : 0.0f;  // triu(k=1) mask
        }
    }

    // ---------- chi-term: electron-nucleus (this lane = electron `lane`) ----------
    const float rn[8] = {nv0.x, nv0.y, nv0.z, nv0.w, nv1.x, nv1.y, nv1.z, nv1.w};
    float chi_acc = 0.0f;
    #pragma unroll
    for (int m = 0; m < N_NUC; ++m) {
        const int    tix = type_idx[m];     // wave-uniform gathers -> SGPRs
        const float  Lm  = L_chi[tix];
        const float  Zm  = Z[m];
        const float* bt  = beta + tix * 8;
        const float  b0  = bt[0];
        // -Z/(-L)^3 + b0*3/L  ==  Z/L^3 + 3*b0/L
        const float  lin = Zm / (Lm * Lm * Lm) + 3.0f * b0 / Lm;
        const float  r   = rn[m];
        float h = bt[7];
        h = fmaf(h, r, bt[6]);
        h = fmaf(h, r, bt[5]);
        h = fmaf(h, r, bt[4]);
        h = fmaf(h, r, bt[3]);
        h = fmaf(h, r, bt[2]);
        h = fmaf(h, r, bt[1]);
        h = fmaf(h, r, lin);
        h = fmaf(h, r, b0);
        const float d = r - Lm;
        chi_acc += (r < Lm) ? d * d * d * h : 0.0f;
    }

    // chi: sum over electrons, MEAN over the 8 nuclei
    float partial = fmaf(chi_acc, 1.0f / (float)N_NUC, u_acc);

    // ---------- full-wave reduction on the matrix pipe (f32 WMMA, exact) ----------
    float total;
#if __has_builtin(__builtin_amdgcn_wmma_f32_16x16x4_f32)
    {
        // Stage 1: A = (partials, 0) [16x4 layout: v0 lanes0-15 K=0, lanes16-31 K=2],
        // B = ones -> D[m,n] = p[m] + p[m+16] = q[m] replicated across columns.
        v2f av;   av[0] = partial; av[1] = 0.0f;
        v2f ones; ones[0] = 1.0f;  ones[1] = 1.0f;
        v8f d1 = {0.0f, 0.0f, 0.0f, 0.0f, 0.0f, 0.0f, 0.0f, 0.0f};
        d1 = __builtin_amdgcn_wmma_f32_16x16x4_f32(
                 false, av, false, ones, (short)0, d1, false, false);
        // Stage 2: feed D's VGPR pairs back as A (A[m,k] picks up q[2p],q[2p+1],
        // q[2p+8],q[2p+9] per the 16x4 layout); accumulate over the 4 pairs.
        v8f tot = {0.0f, 0.0f, 0.0f, 0.0f, 0.0f, 0.0f, 0.0f, 0.0f};
        #pragma unroll
        for (int p = 0; p < 4; ++p) {
            v2f ap; ap[0] = d1[2 * p]; ap[1] = d1[2 * p + 1];
            tot = __builtin_amdgcn_wmma_f32_16x16x4_f32(
                      false, ap, false, ones, (short)0, tot, false, false);
        }
        total = tot[0];   // every element of tot == sum of all 32 partials
    }
#else
    total = partial;
    #pragma unroll
    for (int o = 16; o; o >>= 1) total += __shfl_xor(total, o, 32);
#endif

    if (lane == 0) out[b] = total;
}

extern "C" void kernel_launch(void* const* d_in, const int* in_sizes, int n_in,
                              void* d_out, int out_size, void* d_ws, size_t ws_size,
                              hipStream_t stream) {
    const float* elec     = (const float*)d_in[0];
    const float* nuc      = (const float*)d_in[1];
    const float* alpha    = (const float*)d_in[2];
    const float* beta     = (const float*)d_in[3];
    const float* Z        = (const float*)d_in[4];
    const float* L_chi    = (const float*)d_in[5];
    const float* L_u      = (const float*)d_in[6];
    const int*   type_idx = (const int*)d_in[7];
    float* out = (float*)d_out;

    const int Bnum   = in_sizes[0] / (N_ELEC * N_ELEC);   // 4096
    const int blocks = (Bnum + 7) / 8;                    // 8 waves per block
    jastrow_kernel<<<blocks, 256, 0, stream>>>(
        elec, nuc, alpha, beta, Z, L_chi, L_u, type_idx, out, Bnum);
}